// CentralityChannel_42992622633780
// MI455X (gfx1250) — compile-verified
//
#include <hip/hip_runtime.h>

#define N_NODES 50000
#define N_EDGES 800000
#define DIM 64
#define ND (N_NODES * DIM)
#define BN_EPS 1e-5f

typedef __attribute__((ext_vector_type(2))) float v2f;
typedef __attribute__((ext_vector_type(8))) float v8f;

// ---------------------------------------------------------------------------
// Zero the aggregation buffer and the 256-float stats area (S1|S2).
// ---------------------------------------------------------------------------
__global__ __launch_bounds__(256) void gin_zero_kernel(float* __restrict__ agg,
                                                       float* __restrict__ stats) {
    int idx = blockIdx.x * 256 + threadIdx.x;
    if (idx < ND) agg[idx] = 0.0f;
    if (idx < 256) stats[idx] = 0.0f;
}

// ---------------------------------------------------------------------------
// Edge scatter: agg[tgt[e]][d] += h[src[e]][d] * w[e].
// 64 lanes per edge -> one fully coalesced 256B row read per edge.
// ---------------------------------------------------------------------------
__global__ __launch_bounds__(256) void gin_scatter_kernel(
    const float* __restrict__ h, const int* __restrict__ edge_index,
    const float* __restrict__ ew, float* __restrict__ agg) {
    int e = blockIdx.x * 4 + (threadIdx.x >> 6);
    int d = threadIdx.x & 63;
    if (e < N_EDGES) {
        int s = edge_index[e];            // src row
        int t = edge_index[N_EDGES + e];  // tgt row
        float w = ew[e];
        atomicAdd(&agg[t * 64 + d], h[s * 64 + d] * w);
    }
}

// ---------------------------------------------------------------------------
// Fused GIN-combine + GEMM1 + per-column BN statistics.
//   A = (1+eps)*h + agg   (staged in LDS, stride 65 to avoid bank conflicts)
//   out = A @ W + bias ; stats[col] += sum(out_col), stats[64+col] += sum(out_col^2)
// Block = 256 threads = 8 waves = 2 row-tiles x 4 col-tiles of 16x16 (32 rows).
// WMMA: v_wmma_f32_16x16x4_f32, 16 K-steps for K=64, exact fp32.
// ---------------------------------------------------------------------------
__global__ __launch_bounds__(256) void gin_combine_gemm_kernel(
    const float* __restrict__ h, const float* __restrict__ agg,
    const float* __restrict__ W, const float* __restrict__ bias,
    const float* __restrict__ eps, int layer, float* __restrict__ out,
    float* __restrict__ stats) {
    __shared__ float Wl[64 * 65];
    __shared__ float Al[32 * 65];
    __shared__ float bl[64];
    const int tid = threadIdx.x;
    const int row0 = blockIdx.x * 32;
    const float c1p = 1.0f + eps[layer];

    for (int i = tid; i < 64 * 64; i += 256)
        Wl[(i >> 6) * 65 + (i & 63)] = W[i];
    if (tid < 64) bl[tid] = bias[tid];
    for (int i = tid; i < 32 * 64; i += 256) {
        int r = i >> 6, k = i & 63;
        int row = row0 + r;
        float v = 0.0f;
        if (row < N_NODES) v = c1p * h[row * 64 + k] + agg[row * 64 + k];
        Al[r * 65 + k] = v;
    }
    __syncthreads();

    const int lane = tid & 31;
    const int wave = tid >> 5;
    const int rt = wave >> 2;          // row tile 0..1
    const int n0 = (wave & 3) * 16;    // col tile base
    const int m = lane & 15;
    const int koff = (lane >> 4) * 2;  // lanes 16-31 carry K+2,K+3

    v8f c = {0.f, 0.f, 0.f, 0.f, 0.f, 0.f, 0.f, 0.f};
#pragma unroll
    for (int kb = 0; kb < 16; ++kb) {
        const int k = kb * 4 + koff;
        v2f a, b;
        a.x = Al[(rt * 16 + m) * 65 + k];
        a.y = Al[(rt * 16 + m) * 65 + k + 1];
        b.x = Wl[k * 65 + n0 + m];
        b.y = Wl[(k + 1) * 65 + n0 + m];
        c = __builtin_amdgcn_wmma_f32_16x16x4_f32(false, a, false, b, (short)0,
                                                  c, false, false);
    }

    const int col = n0 + m;
    const float bcol = bl[col];
    const int rbase = row0 + rt * 16 + (lane >> 4) * 8;
    float s = 0.f, s2 = 0.f;
#pragma unroll
    for (int j = 0; j < 8; ++j) {
        int row = rbase + j;
        if (row < N_NODES) {
            float v = c[j] + bcol;
            out[row * 64 + col] = v;
            s += v;
            s2 += v * v;
        }
    }
    // lanes l and l+16 hold the same output column -> one xor-16 reduce
    s += __shfl_xor(s, 16, 32);
    s2 += __shfl_xor(s2, 16, 32);
    if (lane < 16) {
        atomicAdd(&stats[col], s);
        atomicAdd(&stats[64 + col], s2);
    }
}

// ---------------------------------------------------------------------------
// Fused BN+ReLU (using statsIn) + GEMM2 + stats for the second BN.
//   A = relu(bn(tin)) ; out = A @ W + bias ; statsOut accumulated.
// ---------------------------------------------------------------------------
__global__ __launch_bounds__(256) void gin_bn_gemm_kernel(
    const float* __restrict__ tin, const float* __restrict__ W,
    const float* __restrict__ bias, const float* __restrict__ gamma,
    const float* __restrict__ beta, const float* __restrict__ statsIn,
    float* __restrict__ out, float* __restrict__ statsOut) {
    __shared__ float Wl[64 * 65];
    __shared__ float Al[32 * 65];
    __shared__ float bl[64];
    __shared__ float scl[64];
    __shared__ float shf[64];
    const int tid = threadIdx.x;
    const int row0 = blockIdx.x * 32;

    for (int i = tid; i < 64 * 64; i += 256)
        Wl[(i >> 6) * 65 + (i & 63)] = W[i];
    if (tid < 64) {
        bl[tid] = bias[tid];
        float sm = statsIn[tid], sq = statsIn[64 + tid];
        float mu = sm * (1.0f / N_NODES);
        float var = sq * (1.0f / N_NODES) - mu * mu;
        float sc = gamma[tid] * rsqrtf(var + BN_EPS);
        scl[tid] = sc;
        shf[tid] = beta[tid] - mu * sc;
    }
    __syncthreads();
    for (int i = tid; i < 32 * 64; i += 256) {
        int r = i >> 6, k = i & 63;
        int row = row0 + r;
        float v = 0.0f;
        if (row < N_NODES)
            v = fmaxf(tin[row * 64 + k] * scl[k] + shf[k], 0.0f);
        Al[r * 65 + k] = v;
    }
    __syncthreads();

    const int lane = tid & 31;
    const int wave = tid >> 5;
    const int rt = wave >> 2;
    const int n0 = (wave & 3) * 16;
    const int m = lane & 15;
    const int koff = (lane >> 4) * 2;

    v8f c = {0.f, 0.f, 0.f, 0.f, 0.f, 0.f, 0.f, 0.f};
#pragma unroll
    for (int kb = 0; kb < 16; ++kb) {
        const int k = kb * 4 + koff;
        v2f a, b;
        a.x = Al[(rt * 16 + m) * 65 + k];
        a.y = Al[(rt * 16 + m) * 65 + k + 1];
        b.x = Wl[k * 65 + n0 + m];
        b.y = Wl[(k + 1) * 65 + n0 + m];
        c = __builtin_amdgcn_wmma_f32_16x16x4_f32(false, a, false, b, (short)0,
                                                  c, false, false);
    }

    const int col = n0 + m;
    const float bcol = bl[col];
    const int rbase = row0 + rt * 16 + (lane >> 4) * 8;
    float s = 0.f, s2 = 0.f;
#pragma unroll
    for (int j = 0; j < 8; ++j) {
        int row = rbase + j;
        if (row < N_NODES) {
            float v = c[j] + bcol;
            out[row * 64 + col] = v;
            s += v;
            s2 += v * v;
        }
    }
    s += __shfl_xor(s, 16, 32);
    s2 += __shfl_xor(s2, 16, 32);
    if (lane < 16) {
        atomicAdd(&statsOut[col], s);
        atomicAdd(&statsOut[64 + col], s2);
    }
}

// ---------------------------------------------------------------------------
// Final BN+ReLU of a layer: h = relu(bn2(t2)).
// ---------------------------------------------------------------------------
__global__ __launch_bounds__(256) void gin_bn_relu_kernel(
    const float* __restrict__ tin, const float* __restrict__ gamma,
    const float* __restrict__ beta, const float* __restrict__ stats,
    float* __restrict__ hout) {
    int idx = blockIdx.x * 256 + threadIdx.x;
    if (idx < ND) {
        int col = idx & 63;
        float sm = stats[col], sq = stats[64 + col];
        float mu = sm * (1.0f / N_NODES);
        float var = sq * (1.0f / N_NODES) - mu * mu;
        float sc = gamma[col] * rsqrtf(var + BN_EPS);
        float sh = beta[col] - mu * sc;
        hout[idx] = fmaxf(tin[idx] * sc + sh, 0.0f);
    }
}

// ---------------------------------------------------------------------------
// Output gather: out[i] = h[mask[i]] (1024 x 64).
// ---------------------------------------------------------------------------
__global__ __launch_bounds__(256) void gin_gather_kernel(
    const float* __restrict__ h, const int* __restrict__ mask,
    float* __restrict__ out) {
    int i = blockIdx.x * 4 + (threadIdx.x >> 6);
    int d = threadIdx.x & 63;
    if (i < 1024) out[i * 64 + d] = h[mask[i] * 64 + d];
}

// ---------------------------------------------------------------------------
extern "C" void kernel_launch(void* const* d_in, const int* in_sizes, int n_in,
                              void* d_out, int out_size, void* d_ws,
                              size_t ws_size, hipStream_t stream) {
    const float* x     = (const float*)d_in[0];
    const int*   eidx  = (const int*)d_in[1];
    const float* ew    = (const float*)d_in[2];
    const int*   mask  = (const int*)d_in[3];
    const float* eps   = (const float*)d_in[4];
    const float* W1    = (const float*)d_in[5];
    const float* b1    = (const float*)d_in[6];
    const float* g1    = (const float*)d_in[7];
    const float* be1   = (const float*)d_in[8];
    const float* W2    = (const float*)d_in[9];
    const float* b2    = (const float*)d_in[10];
    const float* g2    = (const float*)d_in[11];
    const float* be2   = (const float*)d_in[12];
    float* out = (float*)d_out;

    // Workspace layout (floats): H | AGG (reused as T2) | T1 | S1[128] S2[128]
    float* H   = (float*)d_ws;
    float* AGG = H + ND;   // also serves as T2 output of GEMM2
    float* T1  = AGG + ND;
    float* S1  = T1 + ND;
    float* S2  = S1 + 128;

    hipMemcpyAsync(H, x, (size_t)ND * sizeof(float), hipMemcpyDeviceToDevice,
                   stream);

    const int zeroBlocks = (ND + 255) / 256;
    const int scatBlocks = (N_EDGES + 3) / 4;
    const int gemmBlocks = (N_NODES + 31) / 32;

    for (int l = 0; l < 3; ++l) {
        gin_zero_kernel<<<zeroBlocks, 256, 0, stream>>>(AGG, S1);
        gin_scatter_kernel<<<scatBlocks, 256, 0, stream>>>(H, eidx, ew, AGG);
        gin_combine_gemm_kernel<<<gemmBlocks, 256, 0, stream>>>(
            H, AGG, W1 + l * DIM * DIM, b1 + l * DIM, eps, l, T1, S1);
        gin_bn_gemm_kernel<<<gemmBlocks, 256, 0, stream>>>(
            T1, W2 + l * DIM * DIM, b2 + l * DIM, g1 + l * DIM, be1 + l * DIM,
            S1, AGG /*T2*/, S2);
        gin_bn_relu_kernel<<<zeroBlocks, 256, 0, stream>>>(
            AGG, g2 + l * DIM, be2 + l * DIM, S2, H);
    }
    gin_gather_kernel<<<256, 256, 0, stream>>>(H, mask, out);
}